// RCU_21869973471640
// MI455X (gfx1250) — compile-verified
//
#include <hip/hip_runtime.h>

typedef __attribute__((ext_vector_type(2))) float v2f;
typedef __attribute__((ext_vector_type(8))) float v8f;

#define Bc 8
#define Cc 256
#define Tc 16
#define HWc 784
#define NTILES_PER_B 49   /* 784 / 16 */

// Each block: one batch b, one 16-column spatial tile. Recurrent state h
// (256 x 16 fp32) lives in LDS in WMMA B-fragment order, double buffered.
// Each of the 16 waves owns a 16-row slice of W kept in registers as
// A-fragments, and produces a 16x16 output tile per timestep via
// 64 chained V_WMMA_F32_16X16X4_F32 (K = 256). x_t loads are issued before
// the WMMA stream so the matrix pipe hides their latency.
__global__ __launch_bounds__(512, 1)
void rcu_wmma_kernel(const float* __restrict__ x,
                     const float* __restrict__ Wm,
                     float* __restrict__ out)
{
    // B-fragment layout per k-step (K=4): [k][lane][2 dwords], 64 k-steps.
    __shared__ float hf[2][64 * 32 * 2];   // 2 x 16 KB

    const int tid  = threadIdx.x;
    const int lane = tid & 31;
    const int wid  = tid >> 5;             // wave id == M-tile index (0..15)

    const int blk    = blockIdx.x;
    const int b      = blk / NTILES_PER_B;
    const int n_base = (blk % NTILES_PER_B) * 16;

    // ---- A-fragment coordinates (16x4 fp32 A layout):
    // lanes 0-15: M=lane, K = {0,1}; lanes 16-31: M=lane-16, K = {2,3}
    const int mrow  = lane & 15;
    const int khalf = (lane >> 4) * 2;
    const int o_a   = wid * 16 + mrow;

    // Load this wave's 16x256 slice of W into registers once (64 x v2f).
    v2f aW[64];
#pragma unroll
    for (int k = 0; k < 64; ++k) {
        aW[k] = *(const v2f*)(Wm + o_a * Cc + k * 4 + khalf);
    }

    // ---- C/D fragment coordinates (16x16 fp32):
    // reg r: lanes 0-15 -> M=r, N=lane; lanes 16-31 -> M=8+r, N=lane-16
    const int ncol = lane & 15;
    const int mhi  = (lane >> 4) * 8;
    const int s    = n_base + ncol;        // spatial column in [0,784)

    // Base global index of this lane's output rows: r stride is Tc*HWc.
    const int gi0 = ((b * Cc + (wid * 16 + mhi)) * Tc) * HWc + s;
    #define GI(r, t) (gi0 + (r) * (Tc * HWc) + (t) * HWc)

    // ---- t = 0: out <- x0, seed h fragments in LDS ----
#pragma unroll
    for (int r = 0; r < 8; ++r) {
        const int o  = wid * 16 + r + mhi;
        const int gi = GI(r, 0);
        const float v = x[gi];
        out[gi] = v;
        // scatter into B-fragment layout: c = o
        const int kp = o >> 2;
        const int dw = o & 1;
        const int lp = ncol + 16 * ((o >> 1) & 1);
        hf[0][(kp * 32 + lp) * 2 + dw] = v;
    }
    __syncthreads();

    int cur = 0;
    for (int t = 1; t < Tc; ++t) {
        // Issue this timestep's x loads FIRST: the 64-WMMA stream below
        // hides their global-memory latency.
        float xv[8];
#pragma unroll
        for (int r = 0; r < 8; ++r) {
            xv[r] = x[GI(r, t)];
        }
        // Warm L2 with next timestep's tile (global_prefetch_b8).
        if (t + 1 < Tc) {
#pragma unroll
            for (int r = 0; r < 8; ++r) {
                __builtin_prefetch(&x[GI(r, t + 1)], 0, 1);
            }
        }

        // GEMM: 16x16 tile, K=256 as 64 chained fp32 WMMAs from LDS B-frags.
        v8f acc = {};
#pragma unroll
        for (int k = 0; k < 64; ++k) {
            const v2f bfrag = *(const v2f*)&hf[cur][(k * 32 + lane) * 2];
            acc = __builtin_amdgcn_wmma_f32_16x16x4_f32(
                /*neg_a=*/false, aW[k],
                /*neg_b=*/false, bfrag,
                /*c_mod=*/(short)0, acc,
                /*reuse_a=*/false, /*reuse_b=*/false);
        }

        const int nxt = cur ^ 1;
#pragma unroll
        for (int r = 0; r < 8; ++r) {
            const int o  = wid * 16 + r + mhi;
            const float v = (acc[r] + xv[r]) * 0.5f;
            out[GI(r, t)] = v;
            const int kp = o >> 2;
            const int dw = o & 1;
            const int lp = ncol + 16 * ((o >> 1) & 1);
            hf[nxt][(kp * 32 + lp) * 2 + dw] = v;
        }
        __syncthreads();
        cur = nxt;
    }
    #undef GI
}

extern "C" void kernel_launch(void* const* d_in, const int* in_sizes, int n_in,
                              void* d_out, int out_size, void* d_ws, size_t ws_size,
                              hipStream_t stream) {
    const float* x   = (const float*)d_in[0];   // (B, C, T, H, W) fp32
    const float* Wm  = (const float*)d_in[1];   // (C, C) fp32
    float*       out = (float*)d_out;           // (B, C, T, H, W) fp32

    dim3 grid(Bc * NTILES_PER_B);   // 392 blocks
    dim3 block(512);                // 16 waves (wave32)
    rcu_wmma_kernel<<<grid, block, 0, stream>>>(x, Wm, out);
}